// RegionLoss_1Class_reg_14439680049764
// MI455X (gfx1250) — compile-verified
//
#include <hip/hip_runtime.h>
#include <hip/hip_bf16.h>

typedef __attribute__((ext_vector_type(4))) float f4;
typedef __attribute__((ext_vector_type(2))) float v2f;
typedef __attribute__((ext_vector_type(8))) float v8f;

#define A_NUM 5
#define HW 4096
#define B_NUM 128

__constant__ float c_anc[10] = {0.57273f, 0.677385f,
                                1.87446f, 2.06253f,
                                3.33843f, 5.47434f,
                                7.88282f, 3.52778f,
                                9.77052f, 9.16828f};

__device__ __forceinline__ float sigm(float x) {
    return 1.0f / (1.0f + __expf(-x));
}

// One block per (b, a) pair: streams 5 channel planes of 4096 f32 each with
// wide non-temporal b128 loads, accumulates the 5 loss partial sums, and
// writes one partial per term to d_ws[term*640 + blk].
__global__ __launch_bounds__(256) void region_main(const float* __restrict__ pred,
                                                   const float* __restrict__ target,
                                                   float* __restrict__ dws) {
    const int blk = blockIdx.x;
    const int b   = blk / A_NUM;
    const int a   = blk % A_NUM;
    const int tid = threadIdx.x;

    // ---- per-batch ground-truth scalars (redundant per thread, trivial) ----
    const float gx = target[b * 4 + 0] * 64.0f;
    const float gy = target[b * 4 + 1] * 64.0f;
    const float gw = target[b * 4 + 2] * 64.0f;
    const float gh = target[b * 4 + 3] * 64.0f;

    int   best  = 0;
    float bestv = -1.0f;
#pragma unroll
    for (int k = 0; k < A_NUM; ++k) {
        const float aw = c_anc[2 * k], ah = c_anc[2 * k + 1];
        const float inter = fminf(aw, gw) * fminf(ah, gh);
        const float uni   = aw * ah + gw * gh - inter;
        const float r     = inter / uni;
        if (r > bestv) { bestv = r; best = k; }
    }
    const int  gi = (int)floorf(gx);
    const int  gj = (int)floorf(gy);
    const bool anc_is_best = (a == best);

    const float AW  = c_anc[2 * a];
    const float AH  = c_anc[2 * a + 1];
    const float txs = gx - (float)gi;                 // special tx
    const float tys = gy - (float)gj;                 // special ty
    const float tws = __logf(gw / c_anc[2 * best]);   // special tw target
    const float ths = __logf(gh / c_anc[2 * best + 1]);
    const float gxm = gx - 0.5f * gw, gxM = gx + 0.5f * gw;
    const float gym = gy - 0.5f * gh, gyM = gy + 0.5f * gh;
    const float garea = gw * gh;

    const float* base = pred + ((size_t)b * 25 + (size_t)a * 5) * HW;
    const f4* p0 = (const f4*)(base + 0 * HW);
    const f4* p1 = (const f4*)(base + 1 * HW);
    const f4* p2 = (const f4*)(base + 2 * HW);
    const f4* p3 = (const f4*)(base + 3 * HW);
    const f4* p4 = (const f4*)(base + 4 * HW);

    float ax = 0.f, ay = 0.f, aww = 0.f, ahh = 0.f, ac = 0.f;

#pragma unroll
    for (int it = 0; it < 4; ++it) {
        const int v = it * 256 + tid;            // f4 index within plane
        const f4 q0 = __builtin_nontemporal_load(&p0[v]);
        const f4 q1 = __builtin_nontemporal_load(&p1[v]);
        const f4 q2 = __builtin_nontemporal_load(&p2[v]);
        const f4 q3 = __builtin_nontemporal_load(&p3[v]);
        const f4 q4 = __builtin_nontemporal_load(&p4[v]);
        const int cell0 = v * 4;
        const int h  = cell0 >> 6;               // row (4-float chunk never crosses a row)
        const int w0 = cell0 & 63;
#pragma unroll
        for (int j = 0; j < 4; ++j) {
            const float sx  = sigm(q0[j]);
            const float sy  = sigm(q1[j]);
            const float twr = q2[j];
            const float thr = q3[j];
            const float pc  = sigm(q4[j]);
            const int   w   = w0 + j;

            const float bx = sx + (float)w;
            const float by = sy + (float)h;
            const float bw = __expf(twr) * AW;
            const float bh = __expf(thr) * AH;

            const float mx = fminf(bx - 0.5f * bw, gxm);
            const float Mx = fmaxf(bx + 0.5f * bw, gxM);
            const float my = fminf(by - 0.5f * bh, gym);
            const float My = fmaxf(by + 0.5f * bh, gyM);
            const float cw = bw + gw - (Mx - mx);
            const float ch = bh + gh - (My - my);
            const float carea = (cw > 0.f && ch > 0.f) ? cw * ch : 0.f;
            const float uarea = bw * bh + garea - carea;
            const float iou   = carea / uarea;

            const bool sp = anc_is_best && (h == gj) && (w == gi);
            const float tx  = sp ? txs : 0.5f;
            const float ty  = sp ? tys : 0.5f;
            const float twt = sp ? tws : 0.0f;
            const float tht = sp ? ths : 0.0f;
            const float cm2 = sp ? 5.0f : ((iou > 0.6f) ? 0.0f : 1.0f);
            const float tcf = sp ? iou : 0.0f;   // cell_iou == this cell's iou

            const float dx = sx - tx, dy = sy - ty;
            const float dw = twr - twt, dh = thr - tht, dc = pc - tcf;
            ax  += dx * dx;
            ay  += dy * dy;
            aww += dw * dw;
            ahh += dh * dh;
            ac  += cm2 * dc * dc;
        }
    }

    // ---- wave32 shuffle reduction, then cross-wave via LDS (deterministic) ----
#pragma unroll
    for (int off = 16; off > 0; off >>= 1) {
        ax  += __shfl_down(ax,  off, 32);
        ay  += __shfl_down(ay,  off, 32);
        aww += __shfl_down(aww, off, 32);
        ahh += __shfl_down(ahh, off, 32);
        ac  += __shfl_down(ac,  off, 32);
    }
    __shared__ float red[8][5];
    const int wv = tid >> 5, ln = tid & 31;
    if (ln == 0) {
        red[wv][0] = ax; red[wv][1] = ay; red[wv][2] = aww;
        red[wv][3] = ahh; red[wv][4] = ac;
    }
    __syncthreads();
    if (tid == 0) {
#pragma unroll
        for (int j = 0; j < 5; ++j) {
            float s = 0.f;
#pragma unroll
            for (int k = 0; k < 8; ++k) s += red[k][j];
            dws[j * 640 + blk] = s;
        }
    }
}

// Small squared-difference reduction for the hand/uvd term (8064 elements).
__global__ __launch_bounds__(256) void hand_loss(const float* __restrict__ uvd_gt,
                                                 const float* __restrict__ pred_uvd,
                                                 float* __restrict__ dws) {
    float acc = 0.f;
    for (int i = threadIdx.x; i < B_NUM * 63; i += 256) {
        const float d = uvd_gt[i] - pred_uvd[i];
        acc += d * d;
    }
#pragma unroll
    for (int off = 16; off > 0; off >>= 1) acc += __shfl_down(acc, off, 32);
    __shared__ float red[8];
    const int wv = threadIdx.x >> 5;
    if ((threadIdx.x & 31) == 0) red[wv] = acc;
    __syncthreads();
    if (threadIdx.x == 0) {
        float s = 0.f;
#pragma unroll
        for (int k = 0; k < 8; ++k) s += red[k];
        dws[3200] = s;
    }
}

// Final reduction: 5 waves, one per loss term. Each wave sums its 640 f32
// partials on the matrix pipe: D += A(16x4) * ones(4x16) via
// v_wmma_f32_16x16x4_f32 (exact f32 row-sum; B=1 makes B's layout moot).
__global__ __launch_bounds__(160) void final_reduce(const float* __restrict__ dws,
                                                    float* __restrict__ out) {
    const int tid  = threadIdx.x;
    const int term = tid >> 5;       // 0..4
    const int lane = tid & 31;
    const int m  = lane & 15;        // A-matrix row held by this lane
    const int kb = (lane >> 4) * 2;  // lanes 0-15 hold K=0,1; lanes 16-31 hold K=2,3

    v8f acc = {0.f, 0.f, 0.f, 0.f, 0.f, 0.f, 0.f, 0.f};
    v2f ones;  ones.x = 1.0f;  ones.y = 1.0f;
    const float* p = dws + term * 640;

#pragma unroll
    for (int c = 0; c < 10; ++c) {   // 10 chunks x 64 values = 640
        v2f afrag;
        afrag.x = p[c * 64 + m * 4 + kb];
        afrag.y = p[c * 64 + m * 4 + kb + 1];
        acc = __builtin_amdgcn_wmma_f32_16x16x4_f32(
            /*neg_a=*/false, afrag, /*neg_b=*/false, ones,
            /*c_mod=*/(short)0, acc, /*reuse_a=*/false, /*reuse_b=*/false);
    }
    // Collapse 16x16 accumulator: per-lane sum of 8 VGPRs gives the half-row
    // totals (lanes 0-15: M=0..7, lanes 16-31: M=8..15), identical across N.
    const float s = acc[0] + acc[1] + acc[2] + acc[3] +
                    acc[4] + acc[5] + acc[6] + acc[7];
    const float tot = __shfl(s, 0, 32) + __shfl(s, 16, 32);

    __shared__ float sums[5];
    if (lane == 0) sums[term] = tot;
    __syncthreads();

    if (tid == 0) {
        const float lx = 0.5f * sums[0];
        const float ly = 0.5f * sums[1];
        const float lw = 0.5f * sums[2];
        const float lh = 0.5f * sums[3];
        const float lc = 0.5f * sums[4];
        const float lhand = 0.5f * dws[3200];
        out[0] = lx + ly + lw + lh + lc + lhand;
        out[1] = lx;
        out[2] = ly;
        out[3] = lw;
        out[4] = lh;
        out[5] = lc;
        out[6] = lhand;
    }
}

extern "C" void kernel_launch(void* const* d_in, const int* in_sizes, int n_in,
                              void* d_out, int out_size, void* d_ws, size_t ws_size,
                              hipStream_t stream) {
    const float* pred     = (const float*)d_in[0];
    const float* pred_uvd = (const float*)d_in[1];
    const float* target   = (const float*)d_in[2];
    const float* uvd_gt   = (const float*)d_in[3];
    // d_in[4] = train_out (always 1 for the tuple output; out_size == 7)

    float* dws = (float*)d_ws;   // layout: [0..3199] term partials, [3200] hand
    float* out = (float*)d_out;

    region_main <<<B_NUM * A_NUM, 256, 0, stream>>>(pred, target, dws);
    hand_loss   <<<1, 256, 0, stream>>>(uvd_gt, pred_uvd, dws);
    final_reduce<<<1, 160, 0, stream>>>(dws, out);
}